// NLB_57750130262587
// MI455X (gfx1250) — compile-verified
//
#include <hip/hip_runtime.h>
#include <math.h>

typedef float v2f __attribute__((ext_vector_type(2)));
typedef float v8f __attribute__((ext_vector_type(8)));

#define DIM   192
#define C2    96
#define HW    4096
#define BATCH 2
#define NROW  (BATCH * HW)   // 8192 rows of the [N,96] matrices
#define KSTR  100            // padded LDS row stride (conflict-free)

// ---------------------------------------------------------------------------
// Kernel 1: fused Q/K/V 1x1 convs.  m{1,2,3} are [NROW x 96] flat (== BCHW).
// ---------------------------------------------------------------------------
__global__ __launch_bounds__(256)
void nlb_qkv(const float* __restrict__ x,
             const float* __restrict__ w1, const float* __restrict__ b1,
             const float* __restrict__ w2, const float* __restrict__ b2,
             const float* __restrict__ w3, const float* __restrict__ b3,
             float* __restrict__ m1, float* __restrict__ m2, float* __restrict__ m3)
{
    const int hw = blockIdx.x * blockDim.x + threadIdx.x;   // 0..4095
    const int bo = blockIdx.y;                              // b*96 + o
    const int b  = bo / C2, o = bo - b * C2;
    const float* xb = x + (size_t)b * DIM * HW + hw;
    const float* W1 = w1 + o * DIM;
    const float* W2 = w2 + o * DIM;
    const float* W3 = w3 + o * DIM;
    float a1 = 0.f, a2 = 0.f, a3 = 0.f;
#pragma unroll 4
    for (int c = 0; c < DIM; ++c) {
        const float xv = xb[(size_t)c * HW];                // coalesced over hw
        a1 = fmaf(xv, W1[c], a1);                           // weights wave-uniform
        a2 = fmaf(xv, W2[c], a2);
        a3 = fmaf(xv, W3[c], a3);
    }
    const size_t idx = (size_t)bo * HW + hw;
    m1[idx] = a1 + b1[o];
    m2[idx] = a2 + b2[o];
    m3[idx] = a3 + b3[o];
}

// ---------------------------------------------------------------------------
// Kernel 2: flash attention with V_WMMA_F32_16X16X4_F32.
// One wave owns 16 queries; 8 waves/block march over key tiles in lockstep,
// sharing LDS-staged K/V tiles.  S^T = K*Q^T puts queries on lanes (cheap
// online softmax); O^T = V^T * P^T after transposing P through 1KB LDS.
// ---------------------------------------------------------------------------
__global__ __launch_bounds__(256)
void nlb_flash(const float* __restrict__ Q, const float* __restrict__ K,
               const float* __restrict__ V, float* __restrict__ O)
{
    __shared__ float sK[16 * KSTR];
    __shared__ float sV[16 * KSTR];
    __shared__ float sP[8 * 256];

    const int tid  = threadIdx.x;
    const int lane = tid & 31;
    const int wave = tid >> 5;
    const int row  = lane & 15;
    const int hi   = lane >> 4;
    const int qbase = (blockIdx.x * 8 + wave) * 16;
    float* myP = sP + wave * 256;

    // Preload Q^T B-operands: qb[cc] lane l = Q[qbase+(l&15)][4cc + 2hi + {0,1}]
    v2f qb[24];
    {
        const float* qp = Q + (size_t)(qbase + row) * C2 + 2 * hi;
#pragma unroll
        for (int cc = 0; cc < 24; ++cc) qb[cc] = *(const v2f*)(qp + 4 * cc);
    }

    v8f acc[6];
#pragma unroll
    for (int jb = 0; jb < 6; ++jb)
#pragma unroll
        for (int e = 0; e < 8; ++e) acc[jb][e] = 0.f;
    float mrun = -INFINITY;
    float lrun = 0.f;

    for (int kt = 0; kt < NROW; kt += 16) {
        __syncthreads();   // previous tile's LDS reads done
        {   // stage 16x96 K and V tiles (rows contiguous -> coalesced float4)
            const float4* ks = (const float4*)(K + (size_t)kt * C2);
            const float4* vs = (const float4*)(V + (size_t)kt * C2);
            for (int i = tid; i < 384; i += 256) {
                const int r = i / 24, c4 = (i - r * 24) * 4;
                *(float4*)(sK + r * KSTR + c4) = ks[i];
                *(float4*)(sV + r * KSTR + c4) = vs[i];
            }
        }
        __syncthreads();

        // S^T tile (16 keys x 16 queries): 24 chained f32 WMMAs over d=96
        v8f s;
#pragma unroll
        for (int e = 0; e < 8; ++e) s[e] = 0.f;
        {
            const float* kp = sK + row * KSTR + 2 * hi;
#pragma unroll
            for (int cc = 0; cc < 24; ++cc) {
                v2f a = *(const v2f*)(kp + 4 * cc);    // A = K rows
                s = __builtin_amdgcn_wmma_f32_16x16x4_f32(
                        false, a, false, qb[cc], (short)0, s, false, false);
            }
        }

        // Online softmax: query == lane&15 (both lane halves duplicate it)
        float tmax = s[0];
#pragma unroll
        for (int i = 1; i < 8; ++i) tmax = fmaxf(tmax, s[i]);
        tmax = fmaxf(tmax, __shfl_xor(tmax, 16, 32));
        const float nm   = fmaxf(mrun, tmax);
        const float corr = __expf(mrun - nm);
        float p[8], ps = 0.f;
#pragma unroll
        for (int i = 0; i < 8; ++i) { p[i] = __expf(s[i] - nm); ps += p[i]; }
        ps += __shfl_xor(ps, 16, 32);
        lrun = lrun * corr + ps;
        mrun = nm;
#pragma unroll
        for (int jb = 0; jb < 6; ++jb)
#pragma unroll
            for (int e = 0; e < 8; ++e) acc[jb][e] *= corr;

        // Transpose P into B-operand layout via LDS: store as [key][query]
#pragma unroll
        for (int i = 0; i < 8; ++i)
            myP[(i + 8 * hi) * 16 + row] = p[i];
        __syncthreads();   // order stores vs cross-lane loads

        v2f bp[4];
#pragma unroll
        for (int t = 0; t < 4; ++t) {
            bp[t].x = myP[(4 * t + 2 * hi) * 16 + row];
            bp[t].y = myP[(4 * t + 2 * hi + 1) * 16 + row];
        }

        // O^T (16 feat x 16 query) += V^T * P^T : 6 feature tiles x 4 k-chunks
#pragma unroll
        for (int jb = 0; jb < 6; ++jb) {
#pragma unroll
            for (int t = 0; t < 4; ++t) {
                v2f a;
                a.x = sV[(4 * t + 2 * hi)     * KSTR + jb * 16 + row];
                a.y = sV[(4 * t + 2 * hi + 1) * KSTR + jb * 16 + row];
                acc[jb] = __builtin_amdgcn_wmma_f32_16x16x4_f32(
                              false, a, false, bp[t], (short)0, acc[jb], false, false);
            }
        }
    }

    // Normalize and store O[q][j]; C-layout: reg i, lane l -> [q=l&15][jb*16+i+8hi]
    const float inv = 1.f / lrun;
    float* op = O + (size_t)(qbase + row) * C2 + 8 * hi;
#pragma unroll
    for (int jb = 0; jb < 6; ++jb)
#pragma unroll
        for (int i = 0; i < 8; ++i)
            op[jb * 16 + i] = acc[jb][i] * inv;
}

// ---------------------------------------------------------------------------
// Kernel 3: final 96->192 projection + bias + residual
// ---------------------------------------------------------------------------
__global__ __launch_bounds__(256)
void nlb_proj(const float* __restrict__ A, const float* __restrict__ wl,
              const float* __restrict__ bl, const float* __restrict__ x,
              float* __restrict__ out)
{
    const int hw = blockIdx.x * blockDim.x + threadIdx.x;
    const int bo = blockIdx.y;                 // b*192 + o
    const int b  = bo / DIM, o = bo - b * DIM;
    const float* Ab = A + (size_t)b * C2 * HW + hw;
    const float* W  = wl + o * C2;
    float acc = bl[o];
#pragma unroll 4
    for (int c = 0; c < C2; ++c) acc = fmaf(Ab[(size_t)c * HW], W[c], acc);
    const size_t idx = (size_t)bo * HW + hw;
    out[idx] = acc + x[idx];
}

// ---------------------------------------------------------------------------
extern "C" void kernel_launch(void* const* d_in, const int* in_sizes, int n_in,
                              void* d_out, int out_size, void* d_ws, size_t ws_size,
                              hipStream_t stream)
{
    const float* x  = (const float*)d_in[0];
    const float* w1 = (const float*)d_in[1];
    const float* b1 = (const float*)d_in[2];
    const float* w2 = (const float*)d_in[3];
    const float* b2 = (const float*)d_in[4];
    const float* w3 = (const float*)d_in[5];
    const float* b3 = (const float*)d_in[6];
    const float* wl = (const float*)d_in[7];
    const float* bl = (const float*)d_in[8];
    float* out = (float*)d_out;

    const size_t msz = (size_t)NROW * C2;      // 786432 floats per matrix
    float* m1 = (float*)d_ws;
    float* m2 = m1 + msz;
    float* m3 = m2 + msz;
    float* ob = m3 + msz;                      // total ws use: ~12.6 MB

    nlb_qkv<<<dim3(HW / 256, BATCH * C2), 256, 0, stream>>>(
        x, w1, b1, w2, b2, w3, b3, m1, m2, m3);
    nlb_flash<<<dim3(NROW / (16 * 8)), 256, 0, stream>>>(m1, m2, m3, ob);
    nlb_proj<<<dim3(HW / 256, BATCH * DIM), 256, 0, stream>>>(ob, wl, bl, x, out);
}